// Stacked_Filter_22754736734701
// MI455X (gfx1250) — compile-verified
//
#include <hip/hip_runtime.h>

typedef __attribute__((ext_vector_type(2))) float v2f;
typedef __attribute__((ext_vector_type(4))) float f4;
typedef __attribute__((ext_vector_type(8))) float v8f;

#define NN 2048   // N (spectral dim = GEMM K-dim, and M-dim)
#define FF 256    // feature dim
#define BM 64     // block tile M
#define BN 64     // block tile N
#define KT 32     // K tile
#define LDT (BM + 4)   // 68 floats -> 272B row stride (16B aligned)

// d[j] = sum_i c_i/((a_i-b_i)^2/4) * relu((a_i - lam_j)*(lam_j - b_i))
__global__ __launch_bounds__(256)
void diag_combine_kernel(const float* __restrict__ lam,
                         const float* __restrict__ a,
                         const float* __restrict__ b,
                         const float* __restrict__ c,
                         float* __restrict__ d, int n, int k) {
    int j = blockIdx.x * blockDim.x + threadIdx.x;
    if (j >= n) return;
    float l = lam[(size_t)j * n + j];
    float acc = 0.f;
    for (int i = 0; i < k; ++i) {
        float ai = a[i], bi = b[i];
        float rep = fmaxf((ai - l) * (l - bi), 0.f);
        float dd = ai - bi;
        float scale = c[i] / (dd * dd * 0.25f);
        acc += scale * rep;
    }
    d[j] = acc;
}

// C[2048, 256] = op(A)[2048,2048] @ B[2048,256]
// MODE 0: op(A)[r,k] = A[k*NN + r]  (A^T), epilogue: C = dvec[r] * acc   (Z = diag(d) U^T x)
// MODE 1: op(A)[r,k] = A[r*NN + k],        epilogue: C = alpha*xres + acc (out = a x + U Z)
// 128 threads = 4 waves, each wave computes a 32x32 tile (2x2 WMMA accumulators).
// Software-pipelined: next K-tile's global loads are issued before the WMMA block.
template <int MODE>
__global__ __launch_bounds__(128)
void gemm_wmma_f32(const float* __restrict__ A, const float* __restrict__ B,
                   const float* __restrict__ dvec, const float* __restrict__ xres,
                   const float* __restrict__ alpha_p, float* __restrict__ C) {
    __shared__ __align__(16) float As[KT][LDT];
    __shared__ __align__(16) float Bs[KT][LDT];

    const int tid  = threadIdx.x;
    const int w    = tid >> 5;        // wave 0..3
    const int lane = tid & 31;
    const int half = lane >> 4;       // lanes 16-31 hold K+2/K+3 halves
    const int l16  = lane & 15;
    const int wm   = (w & 1) << 5;    // wave M offset (0,32)
    const int wn   = (w >> 1) << 5;   // wave N offset (0,32)
    const int r0   = blockIdx.x * BM;
    const int c0   = blockIdx.y * BN;

    v8f acc[2][2] = {};
    f4 ra[4], rb[4];                  // prefetch registers for next K-tile

    auto load_tiles = [&](int k0) {
#pragma unroll
        for (int p = 0; p < 4; ++p) {
            if (MODE == 0) {
                // op(A)[r,k] = A[k*NN + r]: contiguous along r -> float4 along M
                int kk = (tid >> 4) + p * 8;     // 0..31
                int rr = (tid & 15) << 2;        // 0..60
                ra[p] = *(const f4*)&A[(size_t)(k0 + kk) * NN + r0 + rr];
            } else {
                // op(A)[r,k] = A[r*NN + k]: contiguous along k -> float4 along K
                int rr = (tid >> 3) + p * 16;    // 0..63
                int kk = (tid & 7) << 2;         // 0..28
                ra[p] = *(const f4*)&A[(size_t)(r0 + rr) * NN + k0 + kk];
            }
            int kb = (tid >> 4) + p * 8;         // 0..31
            int nn = (tid & 15) << 2;            // 0..60
            rb[p] = *(const f4*)&B[(size_t)(k0 + kb) * FF + c0 + nn];
        }
    };

    auto stage_tiles = [&]() {
#pragma unroll
        for (int p = 0; p < 4; ++p) {
            if (MODE == 0) {
                int kk = (tid >> 4) + p * 8;
                int rr = (tid & 15) << 2;
                *(f4*)&As[kk][rr] = ra[p];
            } else {
                int rr = (tid >> 3) + p * 16;
                int kk = (tid & 7) << 2;
                As[kk + 0][rr] = ra[p].x;
                As[kk + 1][rr] = ra[p].y;
                As[kk + 2][rr] = ra[p].z;
                As[kk + 3][rr] = ra[p].w;
            }
            int kb = (tid >> 4) + p * 8;
            int nn = (tid & 15) << 2;
            *(f4*)&Bs[kb][nn] = rb[p];
        }
    };

    load_tiles(0);   // prologue

    for (int k0 = 0; k0 < NN; k0 += KT) {
        stage_tiles();
        __syncthreads();

        // issue next tile's global loads now; latency hidden under the WMMAs
        if (k0 + KT < NN) load_tiles(k0 + KT);

        // 8 k-steps of 4, 4 WMMAs each = 32 v_wmma_f32_16x16x4_f32
#pragma unroll
        for (int kk = 0; kk < KT; kk += 4) {
            const int ka = kk + (half << 1);     // half 0 -> K,K+1 ; half 1 -> K+2,K+3
            v2f a0, a1, b0, b1;
            a0.x = As[ka][wm + l16];          a0.y = As[ka + 1][wm + l16];
            a1.x = As[ka][wm + 16 + l16];     a1.y = As[ka + 1][wm + 16 + l16];
            b0.x = Bs[ka][wn + l16];          b0.y = Bs[ka + 1][wn + l16];
            b1.x = Bs[ka][wn + 16 + l16];     b1.y = Bs[ka + 1][wn + 16 + l16];
            acc[0][0] = __builtin_amdgcn_wmma_f32_16x16x4_f32(
                false, a0, false, b0, (short)0, acc[0][0], false, false);
            acc[0][1] = __builtin_amdgcn_wmma_f32_16x16x4_f32(
                false, a0, false, b1, (short)0, acc[0][1], false, false);
            acc[1][0] = __builtin_amdgcn_wmma_f32_16x16x4_f32(
                false, a1, false, b0, (short)0, acc[1][0], false, false);
            acc[1][1] = __builtin_amdgcn_wmma_f32_16x16x4_f32(
                false, a1, false, b1, (short)0, acc[1][1], false, false);
        }
        __syncthreads();
    }

    // epilogue: C/D layout = VGPR v: row (v + 8*half), col l16
    const float alpha = (MODE == 1) ? alpha_p[0] : 0.f;
#pragma unroll
    for (int mi = 0; mi < 2; ++mi) {
#pragma unroll
        for (int ni = 0; ni < 2; ++ni) {
#pragma unroll
            for (int v = 0; v < 8; ++v) {
                int r = r0 + wm + mi * 16 + v + (half << 3);
                int cc = c0 + wn + ni * 16 + l16;
                float val = acc[mi][ni][v];
                if (MODE == 0) {
                    C[(size_t)r * FF + cc] = dvec[r] * val;
                } else {
                    C[(size_t)r * FF + cc] =
                        alpha * xres[(size_t)r * FF + cc] + val;
                }
            }
        }
    }
}

extern "C" void kernel_launch(void* const* d_in, const int* in_sizes, int n_in,
                              void* d_out, int out_size, void* d_ws, size_t ws_size,
                              hipStream_t stream) {
    const float* x     = (const float*)d_in[0];   // [N,F]
    const float* lam   = (const float*)d_in[1];   // [N,N] (diagonal content)
    const float* U     = (const float*)d_in[2];   // [N,N]
    const float* a     = (const float*)d_in[3];   // [K]
    const float* b     = (const float*)d_in[4];   // [K]
    const float* c     = (const float*)d_in[5];   // [K]
    const float* alpha = (const float*)d_in[6];   // [1]
    // d_in[7] = edge_index, unused by the reference math
    const int K = in_sizes[3];

    float* dvec = (float*)d_ws;        // [N]
    float* Z    = dvec + NN;           // [N,F] scratch: diag(d) @ U^T @ x
    float* out  = (float*)d_out;

    diag_combine_kernel<<<NN / 256, 256, 0, stream>>>(lam, a, b, c, dvec, NN, K);

    dim3 grid(NN / BM, FF / BN);       // 32 x 4 = 128 workgroups
    // Z = diag(d) * (U^T @ x)
    gemm_wmma_f32<0><<<grid, 128, 0, stream>>>(U, x, dvec, nullptr, nullptr, Z);
    // out = alpha * x + U @ Z
    gemm_wmma_f32<1><<<grid, 128, 0, stream>>>(U, Z, nullptr, x, alpha, out);
}